// SparseGrid_58935541236550
// MI455X (gfx1250) — compile-verified
//
#include <hip/hip_runtime.h>

#define GRID_R 128
#define SX (GRID_R * GRID_R)
#define SY (GRID_R)
#define SH_DIM 27

typedef float v2f __attribute__((ext_vector_type(2)));
typedef float v8f __attribute__((ext_vector_type(8)));

// ---------------------------------------------------------------------------
// Kernel 1: trilinear gather-interpolate. One thread per point. Memory bound:
// keep gathered tables (links/density/sh/temp) cacheable in L2, stream the
// per-point inputs/outputs with non-temporal hints so they do not thrash L2.
// ---------------------------------------------------------------------------
__global__ __launch_bounds__(256) void sample_kernel(
    const float* __restrict__ points, const int* __restrict__ links,
    const float* __restrict__ density, const float* __restrict__ sh,
    const float* __restrict__ temp, float* __restrict__ out, int n)
{
    int p = blockIdx.x * blockDim.x + threadIdx.x;
    if (p >= n) return;

    // streaming read-once -> non-temporal loads
    float px = __builtin_nontemporal_load(points + 3 * (long)p + 0);
    float py = __builtin_nontemporal_load(points + 3 * (long)p + 1);
    float pz = __builtin_nontemporal_load(points + 3 * (long)p + 2);

    const float scale = (float)(GRID_R - 1);
    float fx = fminf(fmaxf(px, 0.0f), 0.999999f) * scale;
    float fy = fminf(fmaxf(py, 0.0f), 0.999999f) * scale;
    float fz = fminf(fmaxf(pz, 0.0f), 0.999999f) * scale;

    int lx = min((int)fx, GRID_R - 2);
    int ly = min((int)fy, GRID_R - 2);
    int lz = min((int)fz, GRID_R - 2);
    float wx = fx - (float)lx;
    float wy = fy - (float)ly;
    float wz = fz - (float)lz;

    int base = lx * SX + ly * SY + lz;

    float accD = 0.0f, accT = 0.0f;
    float accS[SH_DIM];
#pragma unroll
    for (int f = 0; f < SH_DIM; ++f) accS[f] = 0.0f;

#pragma unroll
    for (int c = 0; c < 8; ++c) {
        const int dx = (c >> 2) & 1, dy = (c >> 1) & 1, dz = c & 1;
        int idx = links[base + dx * SX + dy * SY + dz];
        if (idx >= 0) {
            float wc = (dx ? wx : 1.0f - wx) *
                       (dy ? wy : 1.0f - wy) *
                       (dz ? wz : 1.0f - wz);
            accD += wc * density[idx];
            accT += wc * temp[idx];
            const float* sp = sh + (long)idx * SH_DIM;
#pragma unroll
            for (int f = 0; f < SH_DIM; ++f) accS[f] += wc * sp[f];
        }
    }

    // Output layout: out_d [0,N) | out_sh [N,28N) | out_t [28N,29N) | tv [29N]
    // streaming write-once -> non-temporal stores (preserve L2 for the tables)
    __builtin_nontemporal_store(accD, out + p);
    float* osh = out + (long)n + (long)p * SH_DIM;
#pragma unroll
    for (int f = 0; f < SH_DIM; ++f) __builtin_nontemporal_store(accS[f], osh + f);
    __builtin_nontemporal_store(accT, out + 28L * n + p);
}

// ---------------------------------------------------------------------------
// Kernel 2: total-variation stencil over the (R-1)^3 grid. Wave-wide fp32
// reduction on the matrix pipe: V_WMMA_F32_16X16X4_F32 with B = all-ones
// computes row sums, collapsing 32 lane partials in one op (layout-robust:
// sum over D column 0 equals the sum of all 64 A slots for any lane mapping).
// Then an LDS stage collapses the 8 waves of a block to ONE atomic per block
// per accumulator, cutting hot-address atomic traffic 8x vs per-wave atomics.
// NOTE: no early returns before the WMMA — EXEC must be all ones.
// ---------------------------------------------------------------------------
__global__ __launch_bounds__(256) void tv_kernel(
    const int* __restrict__ links, const float* __restrict__ density,
    float* __restrict__ ws, int ncells)
{
    __shared__ float red[16];               // [0..7] tv partials, [8..15] counts

    int i = blockIdx.x * blockDim.x + threadIdx.x;

    float tvv = 0.0f, cntv = 0.0f;
    if (i < ncells) {
        const int RM = GRID_R - 1;           // 127
        int x = i / (RM * RM);
        int rem = i - x * (RM * RM);
        int y = rem / RM;
        int z = rem - y * RM;

        int base = x * SX + y * SY + z;
        int l000 = links[base];
        int l100 = links[base + SX];
        int l010 = links[base + SY];
        int l001 = links[base + 1];

        if ((l000 | l100 | l010 | l001) >= 0) {   // all four links valid
            float v0 = density[l000];
            float dx = density[l100] - v0;
            float dy = density[l010] - v0;
            float dz = density[l001] - v0;
            tvv  = sqrtf(dx * dx + dy * dy + dz * dz + 1e-12f);
            cntv = 1.0f;
        }
    }

    // ---- wave32 reduction via the matrix unit (B = ones => row-sum) ----
    v2f a1; a1[0] = tvv;  a1[1] = 0.0f;
    v2f a2; a2[0] = cntv; a2[1] = 0.0f;
    v2f b;  b[0] = 1.0f;  b[1] = 1.0f;
    v8f c = {0.0f, 0.0f, 0.0f, 0.0f, 0.0f, 0.0f, 0.0f, 0.0f};

    v8f d1 = __builtin_amdgcn_wmma_f32_16x16x4_f32(false, a1, false, b,
                                                   (short)0, c, false, false);
    v8f d2 = __builtin_amdgcn_wmma_f32_16x16x4_f32(false, a2, false, b,
                                                   (short)0, c, false, false);

    float s1 = d1[0] + d1[1] + d1[2] + d1[3] + d1[4] + d1[5] + d1[6] + d1[7];
    float s2 = d2[0] + d2[1] + d2[2] + d2[3] + d2[4] + d2[5] + d2[6] + d2[7];
    s1 += __shfl_xor(s1, 16, 32);   // combine rows 0-7 and 8-15 halves
    s2 += __shfl_xor(s2, 16, 32);

    int wave = threadIdx.x >> 5;
    int lane = threadIdx.x & 31;
    if (lane == 0) {
        red[wave]     = s1;
        red[8 + wave] = s2;
    }
    __syncthreads();

    if (threadIdx.x == 0) {
        float t1 = 0.0f, t2 = 0.0f;
#pragma unroll
        for (int wv = 0; wv < 8; ++wv) {
            t1 += red[wv];
            t2 += red[8 + wv];
        }
        atomicAdd(&ws[0], t1);      // tv sum
        atomicAdd(&ws[1], t2);      // valid count (exact: integers < 2^23)
    }
}

__global__ void init_ws(float* ws) { ws[0] = 0.0f; ws[1] = 0.0f; }

__global__ void finalize_tv(const float* __restrict__ ws, float* __restrict__ out_tv)
{
    float denom = fmaxf(ws[1], 1.0f);
    *out_tv = ws[0] / denom;
}

// ---------------------------------------------------------------------------
extern "C" void kernel_launch(void* const* d_in, const int* in_sizes, int n_in,
                              void* d_out, int out_size, void* d_ws, size_t ws_size,
                              hipStream_t stream)
{
    const float* points  = (const float*)d_in[0];
    const int*   links   = (const int*)  d_in[1];
    const float* density = (const float*)d_in[2];
    const float* sh      = (const float*)d_in[3];
    const float* temp    = (const float*)d_in[4];
    float* out = (float*)d_out;
    float* ws  = (float*)d_ws;

    const int n = in_sizes[0] / 3;                 // N points
    const int ncells = (GRID_R - 1) * (GRID_R - 1) * (GRID_R - 1);

    init_ws<<<1, 1, 0, stream>>>(ws);

    int blocks = (n + 255) / 256;
    sample_kernel<<<blocks, 256, 0, stream>>>(points, links, density, sh, temp, out, n);

    int tvblocks = (ncells + 255) / 256;
    tv_kernel<<<tvblocks, 256, 0, stream>>>(links, density, ws, ncells);

    finalize_tv<<<1, 1, 0, stream>>>(ws, out + 29L * n);
}